// AssistedExcitation_70420283785392
// MI455X (gfx1250) — compile-verified
//
#include <hip/hip_runtime.h>

typedef float v2f __attribute__((ext_vector_type(2)));
typedef float v8f __attribute__((ext_vector_type(8)));

#define B_   16
#define C_   256
#define H_   128
#define W_   128
#define HW_  (H_ * W_)        // 16384
#define NBOX 320

// -------------------------------------------------------------------------
// Kernel 1: rasterize 320 boxes into a per-image uint8 union mask [B*H*W].
// Box coords precomputed into LDS once per block; each thread owns a pixel.
// -------------------------------------------------------------------------
__global__ __launch_bounds__(256)
void ae_mask_kernel(const float* __restrict__ bboxes,
                    const int*   __restrict__ batch_idx,
                    unsigned char* __restrict__ mask) {
    __shared__ int sx1[NBOX], sy1[NBOX], sx2[NBOX], sy2[NBOX], sb[NBOX];
    const int tid = threadIdx.x;

    for (int n = tid; n < NBOX; n += 256) {
        float xc = bboxes[4 * n + 0];
        float yc = bboxes[4 * n + 1];
        float bw = bboxes[4 * n + 2];
        float bh = bboxes[4 * n + 3];
        // (int) cast truncates toward zero == jnp.trunc + astype(int32)
        int x1 = max(0,      (int)((xc - bw * 0.5f) * (float)W_));
        int y1 = max(0,      (int)((yc - bh * 0.5f) * (float)H_));
        int x2 = min(W_ - 1, (int)((xc + bw * 0.5f) * (float)W_));
        int y2 = min(H_ - 1, (int)((yc + bh * 0.5f) * (float)H_));
        bool valid = (x2 > x1) && (y2 > y1);
        sx1[n] = valid ? x1 : 0x7fffffff;   // invalid box can never match
        sy1[n] = y1;
        sx2[n] = x2;
        sy2[n] = y2;
        sb[n]  = batch_idx[n];
    }
    __syncthreads();

    const int pix = blockIdx.x * 256 + tid;   // linear b*HW + hw
    const int b   = pix / HW_;
    const int hw  = pix - b * HW_;
    const int y   = hw / W_;
    const int x   = hw - y * W_;

    int m = 0;
    for (int n = 0; n < NBOX; ++n) {
        m |= (int)((sb[n] == b) & (x >= sx1[n]) & (x <= sx2[n]) &
                   (y >= sy1[n]) & (y <= sy2[n]));
    }
    mask[pix] = (unsigned char)(m & 1);
}

// -------------------------------------------------------------------------
// Kernel 2: one wave32 per 16 contiguous pixels of one image.
// Phase 1: channel sum via ones^T * X on V_WMMA_F32_16X16X4_F32
//          (64 accumulating WMMAs cover all 256 channels; D columns hold
//           per-pixel channel sums, identical in every row, so any lane
//           reads its pixel's sum from acc[0] with column = lane%16).
// Phase 2: out = x + mask * sum/256. Re-reads x (268MB of fp32)
//          with NT loads (last use) and writes out with NT stores so the
//          write-once output does not evict x from L2.
// -------------------------------------------------------------------------
__global__ __launch_bounds__(256)
void ae_main_kernel(const float* __restrict__ x,
                    const unsigned char* __restrict__ mask,
                    float* __restrict__ out) {
    const int lane   = threadIdx.x & 31;
    const int wave   = blockIdx.x * (256 / 32) + (threadIdx.x >> 5); // 0..16383
    const int b      = wave >> 10;                // 1024 pixel-groups per image
    const int hwBase = (wave & 1023) << 4;        // 16 pixels per wave
    const int hw     = hwBase + (lane & 15);
    const int half   = lane >> 4;                 // 0 or 1

    // Phase 1: lanes 0-15 feed channels {k,k+1}, lanes 16-31 feed {k+2,k+3}
    // for pixel (lane&15). Exact K-slot->VGPR mapping is irrelevant because
    // A is all-ones and every channel is loaded exactly once per pixel.
    const float* px = x + ((size_t)b * C_ + (size_t)(half << 1)) * HW_ + hw;

    v8f acc = {};
    v2f ones = {1.0f, 1.0f};

    #pragma unroll 4
    for (int k = 0; k < C_; k += 4) {
        v2f bvec;
        bvec.x = px[0];
        bvec.y = px[HW_];
        acc = __builtin_amdgcn_wmma_f32_16x16x4_f32(
                  /*neg_a=*/false, ones,
                  /*neg_b=*/false, bvec,
                  /*c_mod=*/(short)0, acc,
                  /*reuse_a=*/false, /*reuse_b=*/false);
        px += 4 * HW_;
    }

    const float sum    = acc[0];                       // column = lane%16
    const float m      = (float)mask[b * HW_ + hw];
    const float excite = m * sum * (1.0f / (float)C_); // ALPHA = 1

    // Phase 2: lanes 0-15 -> even channels, 16-31 -> odd channels.
    // Running pointers + unroll -> immediate-offset addressing (stride
    // 2*HW*4 = 0x20000 bytes fits the signed 24-bit IOFFSET).
    const size_t base = ((size_t)b * C_ + (size_t)half) * HW_ + (size_t)hw;
    const float* pin  = x + base;
    float*       pout = out + base;

    #pragma unroll 4
    for (int c = 0; c < C_; c += 2) {
        float v = __builtin_nontemporal_load(pin);
        __builtin_nontemporal_store(v + excite, pout);
        pin  += 2 * HW_;
        pout += 2 * HW_;
    }
}

// -------------------------------------------------------------------------
extern "C" void kernel_launch(void* const* d_in, const int* in_sizes, int n_in,
                              void* d_out, int out_size, void* d_ws, size_t ws_size,
                              hipStream_t stream) {
    const float* x         = (const float*)d_in[0];
    const float* bboxes    = (const float*)d_in[1];
    const int*   batch_idx = (const int*)d_in[2];
    float*       out       = (float*)d_out;
    unsigned char* mask    = (unsigned char*)d_ws;   // 256 KB used

    // 262144 pixels total
    ae_mask_kernel<<<(B_ * HW_) / 256, 256, 0, stream>>>(bboxes, batch_idx, mask);

    // 16384 waves, 8 waves/block -> 2048 blocks
    ae_main_kernel<<<(B_ * HW_ / 16) / 8, 256, 0, stream>>>(x, mask, out);
}